// IWT_45045617001000
// MI455X (gfx1250) — compile-verified
//
#include <hip/hip_runtime.h>

// Haar inverse wavelet transform on gfx1250 via V_WMMA_F32_16X16X4_F32.
// in : [16,128,128,256] f32  (channels: [x1|x2|x3|x4], 64 each)
// out: [16,256,256,64]  f32
//
// Per pixel: out[c,t] = sum_k x_{k+1}[c] * B[k][t], B = 0.5 * Haar sign matrix.
// A (16x4) = 16 channels x 4 subbands (data), B (4x16) = constant Haar
// (columns = output type t, only t<4 used), D (16x16) = results.
// 4 WMMAs cover the 64 channels of one pixel per wave.

typedef float v2f __attribute__((ext_vector_type(2)));
typedef float v4f __attribute__((ext_vector_type(4)));
typedef float v8f __attribute__((ext_vector_type(8)));

#define OUT_ROWSTRIDE 16384  // 2W * 64 floats per output row (W=128)

__global__ __launch_bounds__(256) void iwt_haar_wmma(const float* __restrict__ in,
                                                     float* __restrict__ out,
                                                     int npix) {
    const int gid  = blockIdx.x * blockDim.x + threadIdx.x;
    const int pix  = gid >> 5;            // one pixel per wave32
    if (pix >= npix) return;              // uniform per wave: EXEC stays all-1s
    const int lane = threadIdx.x & 31;
    const int n    = lane & 15;           // A: row M / B,D: column N
    const bool hi  = lane >= 16;          // K-half selector (A/B), M+8 half (D)

    // ---- constant B (4x16), per-lane: {B[k0][n], B[k0+1][n]}, k0 = hi?2:0 ----
    float b0, b1;
    if (n < 4) {
        if (!hi) {                        // rows k=0: [+ + + +], k=1: [- + - +]
            b0 = 0.5f;
            b1 = (n & 1) ? 0.5f : -0.5f;
        } else {                          // rows k=2: [- - + +], k=3: [+ - - +]
            b0 = (n & 2) ? 0.5f : -0.5f;
            b1 = ((n ^ (n >> 1)) & 1) ? -0.5f : 0.5f;
        }
    } else {
        b0 = 0.0f; b1 = 0.0f;             // unused D columns -> 0
    }
    v2f Bv; Bv.x = b0; Bv.y = b1;

    // ---- A loads: documented 16x4 f32 layout.
    // v0: lanes 0-15 -> K=0 (x1 slab), lanes 16-31 -> K=2 (x3 slab)
    // v1: lanes 0-15 -> K=1 (x2 slab), lanes 16-31 -> K=3 (x4 slab)
    const float* __restrict__ pin = in + (size_t)pix * 256;
    const int offA = n + (hi ? 128 : 0);

    v8f D[4];
#pragma unroll
    for (int j = 0; j < 4; ++j) {
        const int c0 = j * 16;            // channel chunk of this WMMA
        v2f A;
        A.x = pin[c0 + offA];             // x1 / x3
        A.y = pin[c0 + offA + 64];        // x2 / x4
        v8f C = {};
        // (neg_a, A, neg_b, B, c_mod, C, reuse_a, reuse_b)
        D[j] = __builtin_amdgcn_wmma_f32_16x16x4_f32(
            false, A, false, Bv, (short)0, C, false, false);
    }

    // ---- stores: lane t (and 16+t) holds 8 consecutive channels of output
    // type t across D[0..7]. t order: 0=ee(2h,2w) 1=eo(2h,2w+1) 2=oe(2h+1,2w)
    // 3=oo(2h+1,2w+1). Two contiguous 16B NT stores per lane per chunk.
    if (n < 4) {
        const int w  = pix & 127;         // W = 128
        const int hb = pix >> 7;          // b*128 + h
        // rowE base = 32768*hb + 128*w  (== b*256*256*64 + 2h*16384 + 2w*64)
        size_t base = (size_t)hb * 32768 + (size_t)w * 128;
        base += (n & 2) ? (size_t)OUT_ROWSTRIDE : 0;  // rowO for t=2,3
        base += (n & 1) ? 64 : 0;                     // +1 pixel for t=1,3
        base += hi ? 8 : 0;                           // channel half
#pragma unroll
        for (int j = 0; j < 4; ++j) {
            float* p = out + base + (size_t)j * 16;
            v4f lo4 = __builtin_shufflevector(D[j], D[j], 0, 1, 2, 3);
            v4f hi4 = __builtin_shufflevector(D[j], D[j], 4, 5, 6, 7);
            __builtin_nontemporal_store(lo4, (v4f*)p);
            __builtin_nontemporal_store(hi4, (v4f*)(p + 4));
        }
    }
}

extern "C" void kernel_launch(void* const* d_in, const int* in_sizes, int n_in,
                              void* d_out, int out_size, void* d_ws, size_t ws_size,
                              hipStream_t stream) {
    (void)d_ws; (void)ws_size; (void)out_size; (void)n_in;
    const float* in  = (const float*)d_in[0];
    float*       out = (float*)d_out;
    const int npix    = in_sizes[0] / 256;       // B*H*W = 262144
    const int threads = 256;                     // 8 waves/block
    const int blocks  = (npix * 32 + threads - 1) / threads;
    iwt_haar_wmma<<<blocks, threads, 0, stream>>>(in, out, npix);
}